// DynamicGraphBlock_1365799600614
// MI455X (gfx1250) — compile-verified
//
#include <hip/hip_runtime.h>
#include <math.h>

// ---- problem constants (fixed by reference) ----
#define B_ 16
#define C_ 128
#define N_ 1024      // H*W = 32*32
#define K_ 9
#define NEG_SLOPE 0.2f
#define BN_EPS 1e-5f

typedef float v2f __attribute__((ext_vector_type(2)));
typedef float v8f __attribute__((ext_vector_type(8)));

// CDNA5 fp32 matrix core: D = A(16x4) * B(4x16) + C(16x16)
__device__ __forceinline__ v8f wmma_f32(v2f a, v2f b, v8f c) {
  return __builtin_amdgcn_wmma_f32_16x16x4_f32(
      /*neg_a=*/false, a, /*neg_b=*/false, b,
      /*c_mod=*/(short)0, c, /*reuse_a=*/false, /*reuse_b=*/false);
}

// ---------------------------------------------------------------------------
// Kernel 0: per-point squared norm  sq[b][n] = sum_c x[b][c][n]^2
// ---------------------------------------------------------------------------
__global__ __launch_bounds__(256) void k_sq(const float* __restrict__ x,
                                            float* __restrict__ sqbuf) {
  int i = blockIdx.x * 256 + threadIdx.x;   // i < B_*N_
  int b = i >> 10, n = i & (N_ - 1);
  const float* p = x + b * C_ * N_ + n;
  float s = 0.f;
#pragma unroll 8
  for (int c = 0; c < C_; c++) { float v = p[c * N_]; s += v * v; }
  sqbuf[i] = s;
}

// ---------------------------------------------------------------------------
// Kernel 1: Gram row-tile (16 x 1024) via WMMA f32 + fused top-9 selection.
// One workgroup = one (batch, 16-row tile). 8 waves, wave w owns 128 columns.
// val[n][m] = sq[m] - 2*G[n][m]  (same ordering as d2 per row; self is min).
// ---------------------------------------------------------------------------
__global__ __launch_bounds__(256) void k_gram_topk(const float* __restrict__ x,
                                                   const float* __restrict__ sqbuf,
                                                   int* __restrict__ idxws) {
  __shared__ float vals[16 * 1026];          // 16 rows, padded stride
  const int tid  = threadIdx.x;
  const int lane = tid & 31, wv = tid >> 5;
  const int half = lane >> 4, lrow = lane & 15;
  const int b  = blockIdx.x >> 6;
  const int r0 = (blockIdx.x & 63) << 4;
  const float* xb = x + b * C_ * N_;

  v8f acc[8];
#pragma unroll
  for (int t = 0; t < 8; t++) acc[t] = (v8f)0.f;

  const int colw = wv << 7;                  // this wave's 128-column window
  for (int k = 0; k < 32; k++) {             // K = 128, 4 per step
    int c0 = 4 * k + 2 * half;
    v2f a;
    a.x = xb[c0 * N_ + r0 + lrow];           // A[m][k]   = feats[r0+m][k] = x[b][k][r0+m]
    a.y = xb[(c0 + 1) * N_ + r0 + lrow];
#pragma unroll
    for (int t = 0; t < 8; t++) {
      int cb = colw + (t << 4);
      v2f bb;
      bb.x = xb[c0 * N_ + cb + lrow];        // B[k][n]   = feats[cb+n][k] = x[b][k][cb+n]
      bb.y = xb[(c0 + 1) * N_ + cb + lrow];
      acc[t] = wmma_f32(a, bb, acc[t]);
    }
  }

  // epilogue: val = sq[col] - 2*G, scatter C/D layout into LDS row-tile
#pragma unroll
  for (int t = 0; t < 8; t++) {
    int cb = colw + (t << 4);
    int col = cb + lrow;
    float sqc = sqbuf[b * N_ + col];
#pragma unroll
    for (int v = 0; v < 8; v++) {
      int M = v + 8 * half;                  // C/D: lane<16 -> M=v, lane>=16 -> M=v+8
      vals[M * 1026 + col] = sqc - 2.0f * acc[t][v];
    }
  }
  __syncthreads();

  // top-9 smallest per row: wave handles 2 rows, 9 argmin passes with shuffles
  for (int s = 0; s < 2; s++) {
    int rr = 2 * wv + s;
    volatile float* vrow = vals + rr * 1026;
    int n = r0 + rr;
    for (int kk = 0; kk < K_; kk++) {
      float best = 3.4e38f; int bi = N_;
      for (int j = lane; j < N_; j += 32) {
        float vv = vrow[j];
        if (vv < best) { best = vv; bi = j; }
      }
#pragma unroll
      for (int off = 16; off > 0; off >>= 1) {
        float ov = __shfl_xor(best, off, 32);
        int   oi = __shfl_xor(bi, off, 32);
        if (ov < best || (ov == best && oi < bi)) { best = ov; bi = oi; }
      }
      if (lane == 0) {
        idxws[(b * N_ + n) * K_ + kk] = bi;
        vrow[bi] = 3.4e38f;                  // knock out, next pass finds next-min
      }
    }
  }
}

// ---------------------------------------------------------------------------
// Kernel 2: EdgeConv MLP. One workgroup = one (batch, 16-point tile).
// For each of K=9 neighbor slots: gather nbr tile, GEMM1 (K=256: center half
// from Fc, diff half = Nb-Fc on the fly), +b1, leaky, GEMM2 (K=128), running
// max in registers. All GEMMs on v_wmma_f32_16x16x4_f32. 8 waves = 8 N-tiles.
// ---------------------------------------------------------------------------
__global__ __launch_bounds__(256) void k_edgemlp(const float* __restrict__ x,
                                                 const float* __restrict__ W1,
                                                 const float* __restrict__ b1,
                                                 const float* __restrict__ W2,
                                                 const float* __restrict__ b2,
                                                 const int* __restrict__ idxws,
                                                 float* __restrict__ msg) {
  __shared__ float Fc[16 * 130];             // center features  [p][c]
  __shared__ float Nb[16 * 130];             // neighbor features[p][c]
  __shared__ float H1[16 * 130];             // hidden layer     [p][c]
  __shared__ int   sIdx[16 * K_];

  const int tid  = threadIdx.x;
  const int lane = tid & 31, wv = tid >> 5;
  const int half = lane >> 4, lrow = lane & 15;
  const int b  = blockIdx.x >> 6;
  const int n0 = (blockIdx.x & 63) << 4;
  const float* xb = x + b * C_ * N_;

  for (int i = tid; i < 16 * C_; i += 256) { // coalesced: lanes -> consecutive p
    int p = i & 15, c = i >> 4;
    Fc[p * 130 + c] = xb[c * N_ + n0 + p];
  }
  if (tid < 16 * K_) {
    int p = tid / K_, kk = tid % K_;
    sIdx[tid] = idxws[(b * N_ + n0 + p) * K_ + kk];
  }
  __syncthreads();

  const int nb = wv << 4;                    // this wave's output-column tile
  v8f accmax = (v8f)(-3.4e38f);
  const float b1v = b1[nb + lrow];

  for (int ks = 0; ks < K_; ks++) {
    // gather neighbor feature tile
    for (int i = tid; i < 16 * C_; i += 256) {
      int p = i & 15, c = i >> 4;
      Nb[p * 130 + c] = xb[c * N_ + sIdx[p * K_ + ks]];
    }
    __syncthreads();

    // GEMM1: (16 x 256) . (256 x 128), this wave's 16 columns
    v8f acc1 = (v8f)0.f;
#pragma unroll 4
    for (int t = 0; t < 64; t++) {
      int c0 = 4 * t + 2 * half;
      v2f a, bb;
      if (c0 < C_) {                         // edge[:, :128] = center
        a.x = Fc[lrow * 130 + c0];
        a.y = Fc[lrow * 130 + c0 + 1];
      } else {                               // edge[:, 128:] = nbr - center
        int cc = c0 - C_;
        a.x = Nb[lrow * 130 + cc]     - Fc[lrow * 130 + cc];
        a.y = Nb[lrow * 130 + cc + 1] - Fc[lrow * 130 + cc + 1];
      }
      bb.x = W1[c0 * C_ + nb + lrow];
      bb.y = W1[(c0 + 1) * C_ + nb + lrow];
      acc1 = wmma_f32(a, bb, acc1);
    }
    // bias + leaky ReLU, write hidden tile (writes are row-contiguous)
#pragma unroll
    for (int v = 0; v < 8; v++) {
      float h = acc1[v] + b1v;
      h = (h >= 0.f) ? h : NEG_SLOPE * h;
      H1[(v + 8 * half) * 130 + nb + lrow] = h;
    }
    __syncthreads();

    // GEMM2: (16 x 128) . (128 x 128)
    v8f acc2 = (v8f)0.f;
#pragma unroll 4
    for (int t = 0; t < 32; t++) {
      int c0 = 4 * t + 2 * half;
      v2f a, bb;
      a.x = H1[lrow * 130 + c0];
      a.y = H1[lrow * 130 + c0 + 1];
      bb.x = W2[c0 * C_ + nb + lrow];
      bb.y = W2[(c0 + 1) * C_ + nb + lrow];
      acc2 = wmma_f32(a, bb, acc2);
    }
#pragma unroll
    for (int v = 0; v < 8; v++) accmax[v] = fmaxf(accmax[v], acc2[v]);
    __syncthreads();                         // before Nb/H1 are overwritten
  }

  const float b2v = b2[nb + lrow];
#pragma unroll
  for (int v = 0; v < 8; v++) {
    int M = v + 8 * half;                    // point within tile
    msg[b * C_ * N_ + (nb + lrow) * N_ + n0 + M] = accmax[v] + b2v;
  }
}

// ---------------------------------------------------------------------------
// Kernel 3: per-channel BN stats over (B, N)
// ---------------------------------------------------------------------------
__global__ __launch_bounds__(256) void k_bnstats(const float* __restrict__ msg,
                                                 float* __restrict__ meanbuf,
                                                 float* __restrict__ invstdbuf) {
  __shared__ float s1[256], s2[256];
  const int c = blockIdx.x, tid = threadIdx.x;
  float sum = 0.f, sq = 0.f;
  for (int i = tid; i < B_ * N_; i += 256) {
    int b = i >> 10, n = i & (N_ - 1);
    float v = msg[b * C_ * N_ + c * N_ + n];
    sum += v; sq += v * v;
  }
  s1[tid] = sum; s2[tid] = sq;
  __syncthreads();
  for (int off = 128; off > 0; off >>= 1) {
    if (tid < off) { s1[tid] += s1[tid + off]; s2[tid] += s2[tid + off]; }
    __syncthreads();
  }
  if (tid == 0) {
    float m = s1[0] / (float)(B_ * N_);
    float var = s2[0] / (float)(B_ * N_) - m * m;
    meanbuf[c] = m;
    invstdbuf[c] = rsqrtf(var + BN_EPS);
  }
}

// ---------------------------------------------------------------------------
// Kernel 4: BN apply + residual + ReLU
// ---------------------------------------------------------------------------
__global__ __launch_bounds__(256) void k_bnapply(const float* __restrict__ msg,
                                                 const float* __restrict__ x,
                                                 const float* __restrict__ gamma,
                                                 const float* __restrict__ beta,
                                                 const float* __restrict__ meanbuf,
                                                 const float* __restrict__ invstdbuf,
                                                 float* __restrict__ out) {
  int i = blockIdx.x * 256 + threadIdx.x;    // flat (B, C, N)
  int c = (i >> 10) & (C_ - 1);
  float v = msg[i];
  float y = gamma[c] * (v - meanbuf[c]) * invstdbuf[c] + beta[c] + x[i];
  out[i] = fmaxf(y, 0.f);
}

// ---------------------------------------------------------------------------
extern "C" void kernel_launch(void* const* d_in, const int* in_sizes, int n_in,
                              void* d_out, int out_size, void* d_ws, size_t ws_size,
                              hipStream_t stream) {
  (void)in_sizes; (void)n_in; (void)out_size; (void)ws_size;
  const float* x     = (const float*)d_in[0];
  const float* W1    = (const float*)d_in[1];
  const float* b1    = (const float*)d_in[2];
  const float* W2    = (const float*)d_in[3];
  const float* b2    = (const float*)d_in[4];
  const float* gamma = (const float*)d_in[5];
  const float* beta  = (const float*)d_in[6];
  float* out = (float*)d_out;

  // workspace layout (floats)
  float* ws        = (float*)d_ws;
  float* sqbuf     = ws;                         // B*N          = 16384
  float* meanbuf   = ws + 16384;                 // C            = 128
  float* invstdbuf = ws + 16512;                 // C            = 128
  int*   idxws     = (int*)(ws + 16640);         // B*N*K        = 147456
  float* msg       = ws + 16640 + 147456;        // B*C*N        = 2097152

  k_sq      <<<(B_ * N_) / 256, 256, 0, stream>>>(x, sqbuf);
  k_gram_topk<<<B_ * (N_ / 16), 256, 0, stream>>>(x, sqbuf, idxws);
  k_edgemlp <<<B_ * (N_ / 16), 256, 0, stream>>>(x, W1, b1, W2, b2, idxws, msg);
  k_bnstats <<<C_, 256, 0, stream>>>(msg, meanbuf, invstdbuf);
  k_bnapply <<<(B_ * C_ * N_) / 256, 256, 0, stream>>>(msg, x, gamma, beta,
                                                       meanbuf, invstdbuf, out);
}